// Attention_51737176048123
// MI455X (gfx1250) — compile-verified
//
#include <hip/hip_runtime.h>
#include <hip/hip_bf16.h>
#include <stdint.h>

// ---------------------------------------------------------------------------
// Decode-step GQA attention for MI455X (gfx1250, wave32).
//   B=8, S=1, D=4096, H=32, KVH=8, HD=128, M=4096  (n_rep = 4)
// Memory-bound (~0.97 GB moved, ~1.2 GFLOP @ 23.3 TB/s => ~42 us floor):
//   - d2d memcpy for the 268 MB cache copy-out
//   - LDS-staged GEMVs (each weight element read from HBM exactly once)
//   - WMMA f32 16x16x4 attention with GLOBAL_LOAD_ASYNC_TO_LDS_B128 staging
// ---------------------------------------------------------------------------

typedef float v2f __attribute__((ext_vector_type(2)));
typedef float v8f __attribute__((ext_vector_type(8)));

#define B_    8
#define D_    4096
#define NH    32
#define NKV   8
#define HD_   128
#define M_    4096
#define NREP  4
#define SCALE 0.08838834764831845f /* 1/sqrt(128) */

// --- CDNA5 async copy: HBM -> LDS, tracked with ASYNCcnt (inline asm; no
// builtin covers these). The 24-bit INST_OFFSET is added to BOTH the LDS and
// the global address, so one base register pair can stage a whole tile.
__device__ __forceinline__ void async_b128(void* lds, const void* g) {
    const unsigned l = (unsigned)(uintptr_t)lds;   // flat LDS addr[31:0] = LDS offset
    asm volatile("global_load_async_to_lds_b128 %0, %1, off"
                 :: "v"(l), "v"(g) : "memory");
}
__device__ __forceinline__ void async_b128_off16(void* lds, const void* g) {
    const unsigned l = (unsigned)(uintptr_t)lds;
    asm volatile("global_load_async_to_lds_b128 %0, %1, off offset:16"
                 :: "v"(l), "v"(g) : "memory");
}
__device__ __forceinline__ void async_wait0() {
    asm volatile("s_wait_asynccnt 0x0" ::: "memory");
}
__device__ __forceinline__ void ds_wait0() {
    asm volatile("s_wait_dscnt 0x0" ::: "memory");
}

// ---------------------------------------------------------------------------
// Fused QKV projection + RoPE + KV-cache scatter.
// One thread per output column (6144 = 4096 q | 1024 k | 1024 v), 8 batch
// accumulators per thread -> weights stream from HBM exactly once. x
// (8x4096 f32 = 128KB) staged into LDS via async-to-LDS DMA.
// ---------------------------------------------------------------------------
__global__ __launch_bounds__(256) void qkv_rope_kernel(
    const float* __restrict__ x,
    const float* __restrict__ wq,
    const float* __restrict__ wk,
    const float* __restrict__ wv,
    const float* __restrict__ cosv,
    const float* __restrict__ sinv,
    const int*   __restrict__ pos_p,
    float* __restrict__ q_out,     // [B_][NH*HD_]
    float* __restrict__ kc_out,    // [B_][NKV][M_][HD_] (holds old cache)
    float* __restrict__ vc_out)
{
    extern __shared__ float sx[];  // B_*D_ floats
    const int tid = threadIdx.x;
    {
        const float* g = x  + tid * 4;
        float*       l = sx + tid * 4;
#pragma unroll
        for (int i = 0; i < (B_ * D_) / 1024; ++i)      // 32 async b128 / thread
            async_b128(l + i * 1024, g + i * 1024);
        async_wait0();
    }
    __syncthreads();

    const int col = blockIdx.x * 256 + tid;              // 0..6143; boundaries are
    const bool isQ = col < NH * HD_;                     // multiples of 256 -> no
    const bool isK = !isQ && col < NH * HD_ + NKV * HD_; // intra-wave divergence
    const float* w;
    int ncols, wcol;
    if (isQ)      { w = wq; ncols = NH * HD_;  wcol = col; }
    else if (isK) { w = wk; ncols = NKV * HD_; wcol = col - NH * HD_; }
    else          { w = wv; ncols = NKV * HD_; wcol = col - NH * HD_ - NKV * HD_; }

    float acc[B_];
#pragma unroll
    for (int b = 0; b < B_; ++b) acc[b] = 0.0f;

    for (int d = 0; d < D_; ++d) {
        const float wval = w[(size_t)d * ncols + wcol];
#pragma unroll
        for (int b = 0; b < B_; ++b)
            acc[b] = __builtin_fmaf(sx[b * D_ + d], wval, acc[b]);
    }

    const int hd = wcol & (HD_ - 1);
    if (isQ || isK) {  // uniform per wave32
        const float c = cosv[hd >> 1];
        const float s = sinv[hd >> 1];
#pragma unroll
        for (int b = 0; b < B_; ++b) {
            const float other = __shfl_xor(acc[b], 1, 32);  // RoPE partner column
            acc[b] = (hd & 1) ? __builtin_fmaf(other, s, acc[b] * c)     // imag
                              : __builtin_fmaf(acc[b], c, -(other * s)); // real
        }
    }

    const int pos = pos_p[0];
    if (isQ) {
#pragma unroll
        for (int b = 0; b < B_; ++b) q_out[b * (NH * HD_) + col] = acc[b];
    } else {
        const int kvh = wcol >> 7;
        float* dst = isK ? kc_out : vc_out;
#pragma unroll
        for (int b = 0; b < B_; ++b)
            dst[(((size_t)b * NKV + kvh) * M_ + pos) * HD_ + hd] = acc[b];
    }
}

// ---------------------------------------------------------------------------
// Attention for one (batch, kv-head): 4 query heads vs 4096 cached keys.
// Pass 1: S = K(16x4) x Q^T(4x16) via v_wmma_f32_16x16x4_f32; per-wave K tiles
//         staged with async-to-LDS (coalesced 512B per instruction per wave).
// Softmax per head (wave32 shuffle reductions).
// Pass 2: O = P(16x4) x V(4x16) via WMMA, shared V tiles async-staged.
// WMMA f32 layouts (ISA 7.12.2):
//   A 16x4 : lane L: M=L%16, K = {hi*2, hi*2+1} of the 4-wide K slice
//   B 4x16 : lane L: N=L%16, K = {hi*2, hi*2+1}
//   C/D    : VGPR r, lane L: M = r + hi*8, N = L%16
// sQ is padded to 16 rows (rows 4..15 zero) so B reads are dense: no EXEC
// divergence anywhere around the WMMA inner loops.
// ---------------------------------------------------------------------------
__global__ __launch_bounds__(256) void attn_kernel(
    const float* __restrict__ q,     // [B_][NH*HD_] (roped)
    const float* __restrict__ kc,    // [B_][NKV][M_][HD_] (updated)
    const float* __restrict__ vc,
    const float* __restrict__ mask,  // [B_][NH][1][M_]
    float* __restrict__ aout)        // [B_][NH*HD_]
{
    extern __shared__ float smem[];
    float* sQ   = smem;                  // 16*128 = 2048 (rows 4..15 zero)
    float* sS   = smem + 2048;           // 4*4096 = 16384 (scores -> probs)
    float* sSum = smem + 2048 + NREP*M_; // 16
    float* sV   = sSum + 16;             // 16*128 = 2048 (shared V tile)
    float* sK   = sV + 16 * HD_;         // 8 waves x 16*128 = 16384

    const int b    = blockIdx.x >> 3;
    const int g    = blockIdx.x & 7;
    const int tid  = threadIdx.x;
    const int wave = tid >> 5;
    const int lane = tid & 31;
    const int n    = lane & 15;          // N column (head / dim)
    const int hi   = lane >> 4;          // 0: K {0,1} | 1: K {2,3}

    // Stage Q (4 heads x 128), zero-pad rows 4..15.
    {
        const float* qp = q + (size_t)b * (NH * HD_) + (size_t)(g * NREP) * HD_;
#pragma unroll
        for (int i = tid; i < 16 * HD_; i += 256)
            sQ[i] = (i < NREP * HD_) ? qp[i] : 0.0f;
    }
    __syncthreads();

    // ---------------- pass 1: scores ----------------
    const float* kbase = kc + (size_t)(b * NKV + g) * M_ * HD_;
    float* sKw = sK + wave * (16 * HD_);            // this wave's private tile
    for (int t = 0; t < 32; ++t) {                  // 8 waves x 32 tiles x 16 keys
        const int keyBase = wave * 512 + t * 16;
        // async-stage K tile (16 keys x 128 dims): 16 coalesced b128 ops/lane-set
        ds_wait0();                                  // tile reuse safety
        {
            const float* gk = kbase + (size_t)keyBase * HD_ + lane * 4;
            float*       lk = sKw + lane * 4;
#pragma unroll
            for (int r = 0; r < 16; ++r)
                async_b128(lk + r * HD_, gk + r * HD_);
        }
        async_wait0();

        v8f acc = {};
#pragma unroll 8
        for (int kk = 0; kk < 32; ++kk) {            // 128 dims, 4 per WMMA
            const int c0 = kk * 4 + hi * 2;
            v2f av; av.x = sKw[n * HD_ + c0]; av.y = sKw[n * HD_ + c0 + 1];
            v2f bv; bv.x = sQ [n * HD_ + c0]; bv.y = sQ [n * HD_ + c0 + 1];
            acc = __builtin_amdgcn_wmma_f32_16x16x4_f32(
                false, av, false, bv, (short)0, acc, false, false);
        }
        if (n < NREP) {                              // valid head columns only
            const float* mp = mask + (size_t)(b * NH + g * NREP + n) * M_;
#pragma unroll
            for (int r = 0; r < 8; ++r) {
                const int key = keyBase + r + hi * 8;    // D row M
                sS[n * M_ + key] = acc[r] * SCALE + mp[key];
            }
        }
    }
    __syncthreads();

    // ---------------- softmax: wave w owns head w (w<4) ----------------
    if (wave < NREP) {
        float* row = sS + wave * M_;
        float m = -3.0e38f;
        for (int i = lane; i < M_; i += 32) m = fmaxf(m, row[i]);
#pragma unroll
        for (int off = 16; off > 0; off >>= 1) m = fmaxf(m, __shfl_xor(m, off, 32));
        float l = 0.0f;
        for (int i = lane; i < M_; i += 32) {
            const float p = __expf(row[i] - m);
            row[i] = p;                              // un-normalized probs
            l += p;
        }
#pragma unroll
        for (int off = 16; off > 0; off >>= 1) l += __shfl_xor(l, off, 32);
        if (lane == 0) sSum[wave] = l;
    }
    __syncthreads();

    // ---------------- pass 2: O = P @ V; wave w owns dims [w*16, w*16+16) ----
    const float* vbase = vc + (size_t)(b * NKV + g) * M_ * HD_;
    const int prow = (n & 3) * M_;                   // clamped probs row (dense load)
    v8f oacc = {};
    for (int kt = 0; kt < M_ / 16; ++kt) {
        {   // async-stage shared V tile (16 keys x 128 dims), 2 b128 per thread
            const int row = tid >> 4;                // 0..15
            const int c   = (tid & 15) * 8;          // 0..120
            float*       lv = sV + row * HD_ + c;
            const float* gv = vbase + (size_t)(kt * 16 + row) * HD_ + c;
            async_b128(lv, gv);
            async_b128_off16(lv, gv);                // offset applies to both sides
            async_wait0();
        }
        __syncthreads();
#pragma unroll
        for (int kk = 0; kk < 4; ++kk) {
            const int k0   = kt * 16 + kk * 4;       // global key base of this WMMA
            const int klds = kk * 4 + hi * 2;        // key within tile for this lane
            const float pa = sS[prow + k0 + hi * 2];         // dense LDS load,
            const float pb = sS[prow + k0 + hi * 2 + 1];     // then cndmask to 0
            v2f av; av.x = (n < NREP) ? pa : 0.0f;
                    av.y = (n < NREP) ? pb : 0.0f;
            v2f bv; bv.x = sV[klds * HD_ + wave * 16 + n];
                    bv.y = sV[(klds + 1) * HD_ + wave * 16 + n];
            oacc = __builtin_amdgcn_wmma_f32_16x16x4_f32(
                false, av, false, bv, (short)0, oacc, false, false);
        }
        __syncthreads();
    }
#pragma unroll
    for (int r = 0; r < 8; ++r) {
        const int head = r + hi * 8;                 // D row M = head
        if (head < NREP) {
            const float inv = 1.0f / sSum[head];
            aout[(size_t)b * (NH * HD_) + (size_t)(g * NREP + head) * HD_ +
                 wave * 16 + n] = oacc[r] * inv;
        }
    }
}

// ---------------------------------------------------------------------------
// Output projection: out = attn(8x4096) @ wo(4096x4096). LDS-staged GEMV.
// ---------------------------------------------------------------------------
__global__ __launch_bounds__(256) void oproj_kernel(
    const float* __restrict__ a,    // [B_][D_]
    const float* __restrict__ wo,   // [D_][D_]
    float* __restrict__ out)        // [B_][D_]
{
    extern __shared__ float sa[];
    const int tid = threadIdx.x;
    {
        const float* g = a  + tid * 4;
        float*       l = sa + tid * 4;
#pragma unroll
        for (int i = 0; i < (B_ * D_) / 1024; ++i)
            async_b128(l + i * 1024, g + i * 1024);
        async_wait0();
    }
    __syncthreads();

    const int col = blockIdx.x * 256 + tid;
    float acc[B_];
#pragma unroll
    for (int b = 0; b < B_; ++b) acc[b] = 0.0f;
    for (int d = 0; d < D_; ++d) {
        const float wval = wo[(size_t)d * D_ + col];
#pragma unroll
        for (int b = 0; b < B_; ++b)
            acc[b] = __builtin_fmaf(sa[b * D_ + d], wval, acc[b]);
    }
#pragma unroll
    for (int b = 0; b < B_; ++b) out[b * D_ + col] = acc[b];
}

// ---------------------------------------------------------------------------
extern "C" void kernel_launch(void* const* d_in, const int* in_sizes, int n_in,
                              void* d_out, int out_size, void* d_ws, size_t ws_size,
                              hipStream_t stream) {
    const float* x    = (const float*)d_in[0];
    const float* fc   = (const float*)d_in[1];
    const float* fs   = (const float*)d_in[2];
    const float* mask = (const float*)d_in[3];
    const float* ck   = (const float*)d_in[4];
    const float* cv   = (const float*)d_in[5];
    // d_in[6] = prefill (0: decode path), d_in[8] = cache_indexes (unused)
    const int*   pos  = (const int*)d_in[7];
    const float* wq   = (const float*)d_in[9];
    const float* wk   = (const float*)d_in[10];
    const float* wv   = (const float*)d_in[11];
    const float* wo   = (const float*)d_in[12];

    float* out    = (float*)d_out;                       // [8][4096]
    float* ck_out = out + (size_t)B_ * D_;               // [8][8][4096][128]
    float* cv_out = ck_out + (size_t)B_ * NKV * M_ * HD_;

    float* q_ws = (float*)d_ws;                          // 8 x 4096 roped Q
    float* a_ws = q_ws + (size_t)B_ * NH * HD_;          // 8 x 4096 attn out

    // Bulk cache copy-out (pure HBM bandwidth; qkv kernel then overwrites row pos).
    const size_t cache_bytes = (size_t)B_ * NKV * M_ * HD_ * sizeof(float);
    hipMemcpyAsync(ck_out, ck, cache_bytes, hipMemcpyDeviceToDevice, stream);
    hipMemcpyAsync(cv_out, cv, cache_bytes, hipMemcpyDeviceToDevice, stream);

    qkv_rope_kernel<<<(NH * HD_ + 2 * NKV * HD_) / 256, 256,
                      B_ * D_ * sizeof(float), stream>>>(
        x, wq, wk, wv, fc, fs, pos, q_ws, ck_out, cv_out);

    // LDS: sQ 2048 + sS 16384 + sSum 16 + sV 2048 + sK 16384 = 36880 floats
    const size_t attn_lds = (size_t)(2048 + NREP * M_ + 16 + 16 * HD_ +
                                     8 * 16 * HD_) * sizeof(float);
    attn_kernel<<<B_ * NKV, 256, attn_lds, stream>>>(q_ws, ck_out, cv_out, mask, a_ws);

    oproj_kernel<<<D_ / 256, 256, B_ * D_ * sizeof(float), stream>>>(a_ws, wo, out);
}